// YoloLoss_58128087384141
// MI455X (gfx1250) — compile-verified
//
#include <hip/hip_runtime.h>

typedef float v2f __attribute__((ext_vector_type(2)));
typedef float v8f __attribute__((ext_vector_type(8)));

#define LAMDA_COORD 5.0f
#define LAMDA_NOOB  0.5f

__device__ __forceinline__ float dev_iou(const float* b1, const float* b2) {
  float x1 = b1[0], y1 = b1[1], w1 = b1[2], h1 = b1[3];
  float x2 = b2[0], y2 = b2[1], w2 = b2[2], h2 = b2[3];
  float xl = fmaxf(x1 - 0.5f * w1, x2 - 0.5f * w2);
  float yt = fmaxf(y1 - 0.5f * h1, y2 - 0.5f * h2);
  float xr = fminf(x1 + 0.5f * w1, x2 + 0.5f * w2);
  float yb = fminf(y1 + 0.5f * h1, y2 + 0.5f * h2);
  // Replicate the reference exactly (mask==1 when boxes do NOT overlap).
  float mask  = ((xr < xl) || (yb < yt)) ? 1.0f : 0.0f;
  float inter = mask * (xr - xl) * (yb - yt);
  float a1 = w1 * h1, a2 = w2 * h2;
  return inter / (a1 + a2 - inter);
}

// Wave32 all-lane sum using V_WMMA_F32_16X16X4_F32.
// A(16x4): lane L<16 holds A[L,0] in a[0]; lane L>=16 holds A[L-16,2] in a[0];
// a[1] (K=1 / K=3) is zero.  B = ones(4x16) (layout-independent).  C = 0.
// D[m,n] = rowsum(A,m) = v(lane m) + v(lane m+16), broadcast over n.
// Sum of a lane's 8 D regs = sum over rows 0-7 (lanes<16) or rows 8-15 (lanes>=16);
// one xor-16 shuffle merges the two halves -> full 32-lane sum in every lane.
__device__ __forceinline__ float wave_sum_wmma(float v) {
  v2f a; a[0] = v;    a[1] = 0.0f;
  v2f b; b[0] = 1.0f; b[1] = 1.0f;
  v8f c = {};
  v8f d = __builtin_amdgcn_wmma_f32_16x16x4_f32(
      /*neg_a=*/false, a, /*neg_b=*/false, b,
      /*c_mod=*/(short)0, c, /*reuse_a=*/false, /*reuse_b=*/false);
  float s = ((d[0] + d[1]) + (d[2] + d[3])) + ((d[4] + d[5]) + (d[6] + d[7]));
  s += __shfl_xor(s, 16, 32);
  return s;
}

__global__ __launch_bounds__(256) void yolo_loss_partial(
    const float* __restrict__ pred, const float* __restrict__ targ,
    float* __restrict__ partials, int ncells, float inv_bs) {
  // acc: [total, xy, wh, conf_obj, conf_noobj, class]
  float acc[6] = {0.f, 0.f, 0.f, 0.f, 0.f, 0.f};

  const int stride = gridDim.x * blockDim.x;
  for (int cidx = blockIdx.x * blockDim.x + threadIdx.x; cidx < ncells; cidx += stride) {
    const float* p = pred + (size_t)cidx * 30;
    const float* t = targ + (size_t)cidx * 30;

    // Prefetch the next tile this lane will touch (speculative, dropped if OOB).
    __builtin_prefetch(p + (size_t)stride * 30, 0, 1);
    __builtin_prefetch(t + (size_t)stride * 30, 0, 1);

    // 8-byte aligned (cell*120 B): load as float2 -> global_load_b64.
    float pv[30], tv[30];
    const v2f* p2 = (const v2f*)p;
    const v2f* t2 = (const v2f*)t;
#pragma unroll
    for (int i = 0; i < 15; ++i) {
      v2f a = p2[i]; pv[2 * i] = a[0]; pv[2 * i + 1] = a[1];
      v2f b = t2[i]; tv[2 * i] = b[0]; tv[2 * i + 1] = b[1];
    }

    float obj   = (tv[4] > 0.0f)  ? 1.0f : 0.0f;
    float noobj = (tv[4] == 0.0f) ? 1.0f : 0.0f;

    float i1 = dev_iou(pv, tv);
    float i2 = dev_iou(pv + 5, tv + 5);
    const float* ch = (i1 > i2) ? pv : (pv + 5);

    float dx = ch[0] - tv[0], dy = ch[1] - tv[1];
    float xy = obj * (dx * dx + dy * dy);

    float dw = __builtin_sqrtf(ch[2]) - __builtin_sqrtf(tv[2]);
    float dh = __builtin_sqrtf(ch[3]) - __builtin_sqrtf(tv[3]);
    float wh = obj * (dw * dw + dh * dh);

    float dc    = ch[4] - tv[4];
    float cobj  = obj * dc * dc;

    float d4 = pv[4] - tv[4], d9 = pv[9] - tv[9];
    float cnoobj = noobj * (d4 * d4 + d9 * d9);

    float cls = 0.0f;
#pragma unroll
    for (int i = 10; i < 30; ++i) {
      float dd = pv[i] - tv[i];
      cls += dd * dd;
    }
    cls *= obj;

    acc[1] += xy;
    acc[2] += wh;
    acc[3] += cobj;
    acc[4] += cnoobj;
    acc[5] += cls;
    acc[0] += LAMDA_COORD * xy + LAMDA_COORD * wh + cobj + LAMDA_NOOB * cnoobj + cls;
  }

#pragma unroll
  for (int j = 0; j < 6; ++j) acc[j] *= inv_bs;

  // Wave-level reduction via WMMA (EXEC all-ones here: uniform post-loop code).
  __shared__ float red[8][6];
  const int wid  = threadIdx.x >> 5;
  const int lane = threadIdx.x & 31;

  float s[6];
#pragma unroll
  for (int j = 0; j < 6; ++j) s[j] = wave_sum_wmma(acc[j]);

  if (lane == 0) {
#pragma unroll
    for (int j = 0; j < 6; ++j) red[wid][j] = s[j];
  }
  __syncthreads();

  if (threadIdx.x < 6) {
    float tot = 0.0f;
    const int nw = blockDim.x >> 5;
    for (int w = 0; w < nw; ++w) tot += red[w][threadIdx.x];
    partials[(size_t)blockIdx.x * 6 + threadIdx.x] = tot;
  }
}

__global__ void yolo_loss_final(const float* __restrict__ partials,
                                float* __restrict__ out, int nblocks) {
  int j = threadIdx.x;
  if (j < 6) {
    float t = 0.0f;
    for (int b = 0; b < nblocks; ++b) t += partials[(size_t)b * 6 + j];
    out[j] = t;  // overwrite (d_out is poisoned, never accumulated into)
  }
}

extern "C" void kernel_launch(void* const* d_in, const int* in_sizes, int n_in,
                              void* d_out, int out_size, void* d_ws, size_t ws_size,
                              hipStream_t stream) {
  const float* pred = (const float*)d_in[0];
  const float* targ = (const float*)d_in[1];
  float* out        = (float*)d_out;
  float* partials   = (float*)d_ws;

  const int ncells  = in_sizes[0] / 30;          // B * 7 * 7
  const float inv_bs = 49.0f / (float)ncells;    // 1 / B

  int blocks = 1024;                             // 8K wave32 waves; 24 KB of d_ws
  size_t need = (size_t)blocks * 6 * sizeof(float);
  if (need > ws_size) blocks = (int)(ws_size / (6 * sizeof(float)));
  if (blocks < 1) blocks = 1;

  hipLaunchKernelGGL(yolo_loss_partial, dim3(blocks), dim3(256), 0, stream,
                     pred, targ, partials, ncells, inv_bs);
  hipLaunchKernelGGL(yolo_loss_final, dim3(1), dim3(32), 0, stream,
                     partials, out, blocks);
}